// GNN_56642028700327
// MI455X (gfx1250) — compile-verified
//
#include <hip/hip_runtime.h>
#include <hip/hip_bf16.h>
#include <math.h>

typedef __attribute__((ext_vector_type(2))) float v2f;
typedef __attribute__((ext_vector_type(8))) float v8f;

#define IN_CH 256
#define HID   16
#define OUTC  40

// ---------------- degree / normalization ----------------

__global__ void k_deg_init(float* __restrict__ deg, int N) {
  int n = blockIdx.x * blockDim.x + threadIdx.x;
  if (n < N) deg[n] = 1.0f;
}

__global__ void k_deg_count(const long long* __restrict__ col, float* __restrict__ deg, int E) {
  int e = blockIdx.x * blockDim.x + threadIdx.x;
  if (e < E) atomicAdd(&deg[(int)col[e]], 1.0f);
}

__global__ void k_dinv(float* __restrict__ d, int N) {
  int n = blockIdx.x * blockDim.x + threadIdx.x;
  if (n < N) d[n] = rsqrtf(d[n]);
}

// ---------------- layer 1 GEMM: h1 = x @ W1  (WMMA f32 16x16x4, K=256) ----------------
// One wave computes a 16(nodes) x 16(HID) f32 tile. W1 (256x16 f32 = 16KB) staged in LDS.

__global__ void k_gemm1_wmma(const float* __restrict__ x, const float* __restrict__ W1,
                             float* __restrict__ h1, int N) {
  __shared__ float sW[IN_CH * HID];
  for (int i = threadIdx.x; i < IN_CH * HID; i += blockDim.x) sW[i] = W1[i];
  __syncthreads();

  int wave = (int)((blockIdx.x * blockDim.x + threadIdx.x) >> 5);
  int lane = threadIdx.x & 31;
  int m0 = wave * 16;
  if (m0 >= N) return;            // whole-wave uniform; EXEC stays all-ones for WMMA

  int mr   = lane & 15;           // A: M row index;  B/D: N column index
  int half = lane >> 4;           // K-pair selector per 32-bit 16x4 layout
  const float* __restrict__ xrow = x + (long long)(m0 + mr) * IN_CH;

  v8f acc = {};
#pragma unroll 8
  for (int k0 = 0; k0 < IN_CH; k0 += 4) {
    int ka = k0 + half * 2;
    // A 16x4: lane<16 holds {K=k0,k0+1}, lane>=16 holds {K=k0+2,k0+3} for row mr
    v2f a = *(const v2f*)(xrow + ka);
    // B 4x16: VGPR0 = rows K=k0 (lanes 0-15) / K=k0+2 (lanes 16-31); VGPR1 = K+1
    v2f b;
    b.x = sW[ka * HID + mr];
    b.y = sW[(ka + 1) * HID + mr];
    acc = __builtin_amdgcn_wmma_f32_16x16x4_f32(false, a, false, b, (short)0, acc,
                                                false, false);
  }

  // D layout: VGPR j -> row (m0 + j + half*8), col mr ; stores are naturally coalesced
  float* __restrict__ out = h1 + (long long)m0 * HID;
#pragma unroll
  for (int j = 0; j < 8; ++j)
    out[(j + half * 8) * HID + mr] = acc[j];
}

// ---------------- aggregation, layer 1 (16-dim) ----------------

__global__ void k_agg1_init(const float* __restrict__ h1, const float* __restrict__ dinv,
                            const float* __restrict__ b1, float* __restrict__ agg1, int N) {
  unsigned gid = blockIdx.x * blockDim.x + threadIdx.x;
  unsigned n = gid >> 4; int d = gid & 15;
  if (n < (unsigned)N) {
    float di = dinv[n];
    agg1[gid] = h1[gid] * di * di + b1[d];
  }
}

__global__ void k_agg1_edges(const long long* __restrict__ row, const long long* __restrict__ col,
                             const float* __restrict__ dinv, const float* __restrict__ h1,
                             float* __restrict__ agg1, int E) {
  unsigned gid = blockIdx.x * blockDim.x + threadIdx.x;
  unsigned e = gid >> 4; int d = gid & 15;
  if (e >= (unsigned)E) return;
  int r = (int)row[e], c = (int)col[e];
  float nrm = dinv[r] * dinv[c];
  atomicAdd(&agg1[(long long)c * HID + d], h1[(long long)r * HID + d] * nrm);
}

// ---------------- layer 2 GEMM: t2 = relu(agg1) @ W2  (100000 x 16 x 40) ----------------

__global__ void k_gemm2(const float* __restrict__ agg1, const float* __restrict__ W2,
                        float* __restrict__ t2, int N) {
  __shared__ float sW[HID * OUTC];
  for (int i = threadIdx.x; i < HID * OUTC; i += blockDim.x) sW[i] = W2[i];
  __syncthreads();
  unsigned gid = blockIdx.x * blockDim.x + threadIdx.x;
  unsigned n = gid / OUTC; int c = (int)(gid % OUTC);
  if (n >= (unsigned)N) return;
  const float* __restrict__ hrow = agg1 + (long long)n * HID;
  float acc = 0.0f;
#pragma unroll
  for (int k = 0; k < HID; ++k)
    acc = fmaf(fmaxf(hrow[k], 0.0f), sW[k * OUTC + c], acc);
  t2[(long long)n * OUTC + c] = acc;
}

// ---------------- aggregation, layer 2 (40-dim), agg2 lives in d_out ----------------

__global__ void k_agg2_init(const float* __restrict__ t2, const float* __restrict__ dinv,
                            const float* __restrict__ b2, float* __restrict__ agg2, int N) {
  unsigned gid = blockIdx.x * blockDim.x + threadIdx.x;
  unsigned n = gid / OUTC; int c = (int)(gid % OUTC);
  if (n < (unsigned)N) {
    float di = dinv[n];
    agg2[gid] = t2[gid] * di * di + b2[c];
  }
}

__global__ void k_agg2_edges(const long long* __restrict__ row, const long long* __restrict__ col,
                             const float* __restrict__ dinv, const float* __restrict__ t2,
                             float* __restrict__ agg2, int E) {
  unsigned gid = blockIdx.x * blockDim.x + threadIdx.x;
  unsigned e = gid / OUTC; int c = (int)(gid % OUTC);
  if (e >= (unsigned)E) return;
  int r = (int)row[e], cc = (int)col[e];
  float nrm = dinv[r] * dinv[cc];
  atomicAdd(&agg2[(long long)cc * OUTC + c], t2[(long long)r * OUTC + c] * nrm);
}

// ---------------- log_softmax over 40 classes, in place on d_out ----------------

__global__ void k_logsoftmax(float* __restrict__ out, int N) {
  int n = blockIdx.x * blockDim.x + threadIdx.x;
  if (n >= N) return;
  float* __restrict__ p = out + (long long)n * OUTC;
  float v[OUTC];
  float mx = -INFINITY;
#pragma unroll
  for (int c = 0; c < OUTC; ++c) { v[c] = p[c]; mx = fmaxf(mx, v[c]); }
  float s = 0.0f;
#pragma unroll
  for (int c = 0; c < OUTC; ++c) s += __expf(v[c] - mx);
  float lse = mx + __logf(s);
#pragma unroll
  for (int c = 0; c < OUTC; ++c) p[c] = v[c] - lse;
}

// ---------------- launcher ----------------

extern "C" void kernel_launch(void* const* d_in, const int* in_sizes, int n_in,
                              void* d_out, int out_size, void* d_ws, size_t ws_size,
                              hipStream_t stream) {
  const float*     x  = (const float*)d_in[0];
  const long long* ei = (const long long*)d_in[1];   // int64 edge_index [2, E]
  const float*     W1 = (const float*)d_in[2];
  const float*     b1 = (const float*)d_in[3];
  const float*     W2 = (const float*)d_in[4];
  const float*     b2 = (const float*)d_in[5];

  const int N = in_sizes[0] / IN_CH;   // 100000
  const int E = in_sizes[1] / 2;       // 3200000
  const long long* row = ei;
  const long long* col = ei + E;

  // workspace layout (floats): dinv[N] | h1[N*16] | agg1[N*16] | t2[N*40]
  float* ws   = (float*)d_ws;
  float* dinv = ws;
  float* h1   = dinv + N;
  float* agg1 = h1 + (size_t)N * HID;
  float* t2   = agg1 + (size_t)N * HID;
  float* agg2 = (float*)d_out;

  const int B = 256;
  auto blocks = [](long long t, int b) { return (unsigned)((t + b - 1) / b); };

  // 1) degrees + dinv
  k_deg_init<<<blocks(N, B), B, 0, stream>>>(dinv, N);
  k_deg_count<<<blocks(E, B), B, 0, stream>>>(col, dinv, E);
  k_dinv<<<blocks(N, B), B, 0, stream>>>(dinv, N);

  // 2) layer 1: WMMA GEMM then normalized scatter-add
  {
    long long waves = (N + 15) / 16;
    k_gemm1_wmma<<<blocks(waves * 32, B), B, 0, stream>>>(x, W1, h1, N);
  }
  k_agg1_init<<<blocks((long long)N * HID, B), B, 0, stream>>>(h1, dinv, b1, agg1, N);
  k_agg1_edges<<<blocks((long long)E * HID, B), B, 0, stream>>>(row, col, dinv, h1, agg1, E);

  // 3) layer 2: ReLU fused into GEMM read, then scatter-add into d_out
  k_gemm2<<<blocks((long long)N * OUTC, B), B, 0, stream>>>(agg1, W2, t2, N);
  k_agg2_init<<<blocks((long long)N * OUTC, B), B, 0, stream>>>(t2, dinv, b2, agg2, N);
  k_agg2_edges<<<blocks((long long)E * OUTC, B), B, 0, stream>>>(row, col, dinv, t2, agg2, E);

  // 4) log_softmax in place
  k_logsoftmax<<<blocks(N, B), B, 0, stream>>>(agg2, N);
}